// MedianFilter3D_85126251807211
// MI455X (gfx1250) — compile-verified
//
#include <hip/hip_runtime.h>
#include <cstdint>

// ---------------------------------------------------------------------------
// 3D median filter 3x3x3, zero-pad SAME, f32, volume (D,H,W) = (64,256,256).
// Exact median of 27 via forgetful selection (provably exact for rank 14/27),
// finished with v_med3_num_f32. Input staged to LDS through the CDNA5 async
// DMA path (global_load_async_to_lds_b32 + s_wait_asynccnt) when available.
// ---------------------------------------------------------------------------

#define VOL_D 64
#define VOL_H 256
#define VOL_W 256

#define TX 8
#define TY 8
#define TZ 4
#define HX (TX + 2)
#define HY (TY + 2)
#define HZ (TZ + 2)
#define TILE_N (HX * HY * HZ)   // 10*10*6 = 600 floats

#if defined(__AMDGCN__) && \
    __has_builtin(__builtin_amdgcn_global_load_async_to_lds_b32) && \
    __has_builtin(__builtin_amdgcn_s_wait_asynccnt)
#define USE_ASYNC_LDS 1
#else
#define USE_ASYNC_LDS 0
#endif

// Compare-exchange: a <- min, b <- max (two VALU ops; independent halves are
// VOPD dual-issue candidates).
__device__ __forceinline__ void ce(float& a, float& b) {
  float lo = __builtin_fminf(a, b);
  float hi = __builtin_fmaxf(a, b);
  a = lo;
  b = hi;
}

// Place min of w[0..S-1] at w[0] and max at w[2*(S/2)-1], preserving the
// multiset. Cost ~ 3S/2 compare-exchanges.
template <int S>
__device__ __forceinline__ void minmax_ends(float* w) {
  constexpr int P = S / 2;
#pragma unroll
  for (int i = 0; i < P; ++i) ce(w[2 * i], w[2 * i + 1]);        // pair phase
#pragma unroll
  for (int i = 1; i < P; ++i) ce(w[0], w[2 * i]);                // min chain
  if constexpr (S & 1) ce(w[0], w[S - 1]);
#pragma unroll
  for (int i = 0; i < P - 1; ++i) ce(w[2 * i + 1], w[2 * P - 1]); // max chain
  if constexpr (S & 1) ce(w[S - 1], w[2 * P - 1]);
}

// One forgetful-selection stage: drop current min & max, insert `nxt`.
// Live count goes S -> S-1. All index math is constexpr => pure register moves.
template <int S>
__device__ __forceinline__ void forget_stage(float* w, float nxt) {
  minmax_ends<S>(w);
  constexpr int M = 2 * (S / 2) - 1;  // slot holding the max
#pragma unroll
  for (int i = 0; i < M - 1; ++i) w[i] = w[i + 1];  // squeeze out slot 0
  if constexpr (S & 1) w[M - 1] = w[S - 1];         // keep odd leftover
  w[S - 2] = nxt;
}

__global__ __launch_bounds__(256) void median3d_gfx1250_kernel(
    const float* __restrict__ g, float* __restrict__ out) {
  __shared__ float tile[TILE_N];

  const int tid = threadIdx.x;
  const int x0 = (int)blockIdx.x * TX - 1;
  const int y0 = (int)blockIdx.y * TY - 1;
  const int z0 = (int)blockIdx.z * TZ - 1;

  // Zero-fill tile => implements zero padding for out-of-volume halo.
#pragma unroll
  for (int e = tid; e < TILE_N; e += 256) tile[e] = 0.0f;
  __syncthreads();

  // Stage the (TX+2)x(TY+2)x(TZ+2) halo tile. In-bounds elements via async
  // DMA to LDS (ASYNCcnt path); out-of-bounds lanes keep the zero fill.
  for (int e = tid; e < TILE_N; e += 256) {
    const int tx = e % HX;
    const int r = e / HX;
    const int ty = r % HY;
    const int tz = r / HY;
    const int gx = x0 + tx;
    const int gy = y0 + ty;
    const int gz = z0 + tz;
    const bool inb = ((unsigned)gx < (unsigned)VOL_W) &&
                     ((unsigned)gy < (unsigned)VOL_H) &&
                     ((unsigned)gz < (unsigned)VOL_D);
    if (inb) {
      const size_t gidx = ((size_t)gz * VOL_H + gy) * (size_t)VOL_W + gx;
#if USE_ASYNC_LDS
      // Builtin takes generic int* pointers (per hipcc diagnostic); codegen
      // resolves address spaces. Order: (global src, lds dst, offset, cpol).
      __builtin_amdgcn_global_load_async_to_lds_b32(
          (int*)(g + gidx), (int*)(&tile[e]), /*offset=*/0, /*cpol=*/0);
#else
      tile[e] = g[gidx];
#endif
    }
  }
#if USE_ASYNC_LDS
  __builtin_amdgcn_s_wait_asynccnt(0);
#endif
  __syncthreads();

  // One output per thread.
  const int lx = tid & 7;
  const int ly = (tid >> 3) & 7;
  const int lz = tid >> 6;

  float v[27];
#pragma unroll
  for (int dz = 0; dz < 3; ++dz)
#pragma unroll
    for (int dy = 0; dy < 3; ++dy)
#pragma unroll
      for (int dx = 0; dx < 3; ++dx)
        v[(dz * 3 + dy) * 3 + dx] =
            tile[(lz + dz) * (HX * HY) + (ly + dy) * HX + (lx + dx)];

  // Forgetful selection: exact rank-14-of-27.
  float w[15];
#pragma unroll
  for (int i = 0; i < 15; ++i) w[i] = v[i];
  forget_stage<15>(w, v[15]);
  forget_stage<14>(w, v[16]);
  forget_stage<13>(w, v[17]);
  forget_stage<12>(w, v[18]);
  forget_stage<11>(w, v[19]);
  forget_stage<10>(w, v[20]);
  forget_stage<9>(w, v[21]);
  forget_stage<8>(w, v[22]);
  forget_stage<7>(w, v[23]);
  forget_stage<6>(w, v[24]);
  forget_stage<5>(w, v[25]);
  forget_stage<4>(w, v[26]);
  const float med = __builtin_amdgcn_fmed3f(w[0], w[1], w[2]);

  const int gx = (int)blockIdx.x * TX + lx;
  const int gy = (int)blockIdx.y * TY + ly;
  const int gz = (int)blockIdx.z * TZ + lz;
  out[((size_t)gz * VOL_H + gy) * (size_t)VOL_W + gx] = med;
}

extern "C" void kernel_launch(void* const* d_in, const int* in_sizes, int n_in,
                              void* d_out, int out_size, void* d_ws,
                              size_t ws_size, hipStream_t stream) {
  (void)in_sizes; (void)n_in; (void)d_ws; (void)ws_size; (void)out_size;
  const float* x = (const float*)d_in[0];
  float* out = (float*)d_out;
  dim3 grid(VOL_W / TX, VOL_H / TY, VOL_D / TZ);  // 32 x 32 x 16
  dim3 block(256);
  median3d_gfx1250_kernel<<<grid, block, 0, stream>>>(x, out);
}